// OptNetEq_11690900979745
// MI455X (gfx1250) — compile-verified
//
#include <hip/hip_runtime.h>
#include <hip/hip_bf16.h>
#include <math.h>

#define NFEAT  1024
#define NHID   256
#define NCLS   10
#define NEQ    64
#define QPENf  0.1f
#define NITER  12
#define NBATCH 128

typedef __attribute__((ext_vector_type(2))) float v2f;
typedef __attribute__((ext_vector_type(8))) float v8f;

// ---- LDS layout for the per-batch QP kernel (floats) ----
#define LDA   260                      // padded stride for A (64 x 256); 1040 B rows (16B-aligned)
#define LDSS  68                       // padded stride for S (64 x 64)
#define OFF_A    0
#define OFF_S    (OFF_A + NEQ * LDA)
#define OFF_Z    (OFF_S + NEQ * LDSS)
#define OFF_INVH (OFF_Z + NHID)
#define OFF_TOP  (OFF_INVH + NHID)
#define OFF_RED  (OFF_TOP + NHID)
#define OFF_NU   (OFF_RED + NHID)
#define OFF_RV   (OFF_NU + NEQ)
#define OFF_RE   (OFF_RV + NEQ)
#define SMEM_FLOATS (OFF_RE + NEQ)
#define SMEM_BYTES  (SMEM_FLOATS * sizeof(float))

// ======================================================================
// Kernel A: h = relu(x @ W1^T + b1) using V_WMMA_F32_16X16X4_F32.
// Grid (16 n-tiles, 8 m-tiles), one wave32 per block, one 16x16 C tile.
// A-frag: row = lane%16, K-pair = 2*(lane/16); B-frag symmetric (col = lane%16).
// ======================================================================
__global__ __launch_bounds__(32) void gemm_h_relu(
    const float* __restrict__ x, const float* __restrict__ W1,
    const float* __restrict__ b1, float* __restrict__ h) {
  const int lane = threadIdx.x;
  const int n0 = blockIdx.x * 16;
  const int m0 = blockIdx.y * 16;
  const int lm = lane & 15;
  const int kh = (lane >> 4) * 2;
  const float* __restrict__ xr = x  + (size_t)(m0 + lm) * NFEAT + kh;
  const float* __restrict__ wr = W1 + (size_t)(n0 + lm) * NFEAT + kh;
  v8f acc = {};
#pragma unroll 8
  for (int k0 = 0; k0 < NFEAT; k0 += 4) {
    v2f av = *(const v2f*)(xr + k0);
    v2f bv = *(const v2f*)(wr + k0);
    acc = __builtin_amdgcn_wmma_f32_16x16x4_f32(false, av, false, bv,
                                                (short)0, acc, false, false);
  }
  const int rbase = m0 + 8 * (lane >> 4);   // C/D layout: VGPR j -> row j + 8*(lane/16)
  const int col   = n0 + lm;
  const float bias = b1[col];
#pragma unroll
  for (int j = 0; j < 8; ++j) {
    float v = acc[j] + bias;
    h[(size_t)(rbase + j) * NHID + col] = v > 0.f ? v : 0.f;
  }
}

// ======================================================================
// Per-batch QP helpers
// ======================================================================
__device__ inline float ratf(float a, float da) {
  return (da < 0.f) ? (-a / da) : 1.0e10f;
}

__device__ inline float block_sum(float v, float* red, int tid) {
  red[tid] = v; __syncthreads();
  for (int o = NHID / 2; o > 0; o >>= 1) {
    if (tid < o) red[tid] += red[tid + o];
    __syncthreads();
  }
  float r = red[0]; __syncthreads();
  return r;
}

__device__ inline float block_min(float v, float* red, int tid) {
  red[tid] = v; __syncthreads();
  for (int o = NHID / 2; o > 0; o >>= 1) {
    if (tid < o) red[tid] = fminf(red[tid], red[tid + o]);
    __syncthreads();
  }
  float r = red[0]; __syncthreads();
  return r;
}

// S = (A .* invH) * A^T  (64x64), 16 tiles of 16x16 over K=256, 8 waves x 2 tiles.
// All 256 threads execute (EXEC all ones at every WMMA, uniform control flow).
__device__ inline void build_S_wmma(const float* As, const float* invHs,
                                    float* Ss, int tid) {
  const int lane = tid & 31;
  const int wave = tid >> 5;
  const int lm = lane & 15;
  const int kh = (lane >> 4) * 2;
#pragma unroll
  for (int tt = 0; tt < 2; ++tt) {
    const int t  = wave + tt * 8;
    const int a0 = (t >> 2) * 16;
    const int b0 = (t & 3) * 16;
    const float* ar = As + (a0 + lm) * LDA + kh;
    const float* br = As + (b0 + lm) * LDA + kh;
    v8f acc = {};
#pragma unroll 4
    for (int k0 = 0; k0 < NHID; k0 += 4) {
      v2f av, bv;
      av.x = ar[k0]     * invHs[k0 + kh];
      av.y = ar[k0 + 1] * invHs[k0 + kh + 1];
      bv.x = br[k0];
      bv.y = br[k0 + 1];
      acc = __builtin_amdgcn_wmma_f32_16x16x4_f32(false, av, false, bv,
                                                  (short)0, acc, false, false);
    }
    const int rb  = a0 + 8 * (lane >> 4);
    const int col = b0 + lm;
#pragma unroll
    for (int j = 0; j < 8; ++j) Ss[(rb + j) * LDSS + col] = acc[j];
  }
  __syncthreads();
}

// In-place Cholesky of S (lower triangle), 64x64, block-parallel.
__device__ inline void cholesky64(float* Ss, int tid) {
  for (int k = 0; k < NEQ; ++k) {
    if (tid == 0) Ss[k * LDSS + k] = sqrtf(Ss[k * LDSS + k]);
    __syncthreads();
    const float dinv = 1.0f / Ss[k * LDSS + k];
    if (tid > k && tid < NEQ) Ss[tid * LDSS + k] *= dinv;
    __syncthreads();
    for (int idx = tid; idx < NEQ * NEQ; idx += NHID) {
      const int i = idx >> 6, j = idx & 63;
      if (j > k && j <= i) Ss[i * LDSS + j] -= Ss[i * LDSS + k] * Ss[j * LDSS + k];
    }
    __syncthreads();
  }
}

// Solve K [dz; dnu] = [top; r2] with K = [[diag(Hd),A^T],[A,0]] via Schur:
//   rhs = A*(invH.*top) + res          (res = -r2; res==re in iterations, -1 at init)
//   L L^T dnu = rhs  (wave-0 substitution, RHS in registers, __shfl broadcasts)
//   dz_i = invH_i * (top_i - (A^T dnu)_i)  -> returned per-thread (i = tid)
__device__ inline float schur_solve(const float* As, const float* Ss,
                                    const float* invHs, const float* tops,
                                    const float* res, float* rvs, float* red,
                                    int tid) {
  {
    const int j = tid >> 2, l = tid & 3;
    const float* ar = As + j * LDA;
    float acc = 0.f;
    for (int i = l; i < NHID; i += 4) acc += ar[i] * invHs[i] * tops[i];
    red[tid] = acc;
  }
  __syncthreads();
  if (tid < NEQ)
    rvs[tid] = red[4*tid] + red[4*tid+1] + red[4*tid+2] + red[4*tid+3] + res[tid];
  __syncthreads();

  if (tid < 32) {                       // wave 0: forward/backward substitution
    const int lane = tid;
    float y0 = rvs[lane], y1 = rvs[lane + 32];
    for (int k = 0; k < NEQ; ++k) {     // L y = rhs
      float yk = (k < 32) ? __shfl(y0, k, 32) : __shfl(y1, k - 32, 32);
      yk /= Ss[k * LDSS + k];
      if (lane == (k & 31)) { if (k < 32) y0 = yk; else y1 = yk; }
      if (lane > k)       y0 -= Ss[lane * LDSS + k] * yk;
      if (lane + 32 > k)  y1 -= Ss[(lane + 32) * LDSS + k] * yk;
    }
    for (int k = NEQ - 1; k >= 0; --k) { // L^T x = y
      float xk = ((k < 32) ? __shfl(y0, k, 32) : __shfl(y1, k - 32, 32))
                 / Ss[k * LDSS + k];
      if (lane == (k & 31)) { if (k < 32) y0 = xk; else y1 = xk; }
      if (lane < k)       y0 -= Ss[k * LDSS + lane] * xk;
      if (lane + 32 < k)  y1 -= Ss[k * LDSS + lane + 32] * xk;
    }
    rvs[lane] = y0; rvs[lane + 32] = y1;
  }
  __syncthreads();

  float at = 0.f;
  for (int j = 0; j < NEQ; ++j) at += As[j * LDA + tid] * rvs[j];
  return invHs[tid] * (tops[tid] - at);
}

// ======================================================================
// Kernel B: one block per batch element. Full interior-point QP in LDS,
// then classifier head + log_softmax. A is staged global->LDS with the
// CDNA5 async copy engine (no VGPR round-trip, overlaps the h load).
// ======================================================================
__global__ __launch_bounds__(NHID) void qp_head(
    const float* __restrict__ h, const float* __restrict__ Ag,
    const float* __restrict__ W2, const float* __restrict__ b2,
    float* __restrict__ out) {
  extern __shared__ float smem[];
  float* As    = smem + OFF_A;
  float* Ss    = smem + OFF_S;
  float* zs    = smem + OFF_Z;
  float* invHs = smem + OFF_INVH;
  float* tops  = smem + OFF_TOP;
  float* red   = smem + OFF_RED;
  float* nus   = smem + OFF_NU;
  float* rvs   = smem + OFF_RV;
  float* res   = smem + OFF_RE;

  const int tid = threadIdx.x;
  const int b   = blockIdx.x;

  // ---- async-stage A (64 KB = 4096 x 16B) into LDS; 16 chunks/thread ----
#pragma unroll 4
  for (int c = 0; c < 16; ++c) {
    const int idx  = (c << 8) | tid;             // 0..4095
    const int row  = idx >> 6;                   // 64 rows
    const int col4 = (idx & 63) << 2;            // float column, 16B aligned
    const unsigned lds_off =                     // flat ptr low 32b == LDS addr
        (unsigned)(size_t)(const void*)(As + row * LDA + col4);
    const unsigned long long gaddr =
        (unsigned long long)(size_t)(const void*)(Ag + (row << 8) + col4);
    asm volatile("global_load_async_to_lds_b128 %0, %1, off"
                 :: "v"(lds_off), "v"(gaddr) : "memory");
  }

  const float p_i = -h[(size_t)b * NHID + tid];  // p = -h (overlaps async copy)

  // ---- initial KKT solve: Hd = QPEN, top = h, r2 = b_eq = 1 (res = -1) ----
  invHs[tid] = 1.0f / QPENf;
  tops[tid]  = -p_i;
  if (tid < NEQ) res[tid] = -1.0f;
#if __has_builtin(__builtin_amdgcn_s_wait_asynccnt)
  __builtin_amdgcn_s_wait_asynccnt(0);
#else
  asm volatile("s_wait_asynccnt 0" ::: "memory");
#endif
  __syncthreads();
  build_S_wmma(As, invHs, Ss, tid);
  cholesky64(Ss, tid);
  float z_i = schur_solve(As, Ss, invHs, tops, res, rvs, red, tid);
  zs[tid] = z_i;
  if (tid < NEQ) nus[tid] = rvs[tid];
  float s_i   = fmaxf(z_i, 1.0f);
  float lam_i = 1.0f;
  __syncthreads();

#pragma unroll 1
  for (int it = 0; it < NITER; ++it) {
    float at = 0.f;                                      // (A^T nu)_i
    for (int j = 0; j < NEQ; ++j) at += As[j * LDA + tid] * nus[j];
    const float rx = QPENf * z_i + p_i - lam_i + at;
    const float ri = s_i - z_i;
    const float D  = lam_i / s_i;
    invHs[tid] = 1.0f / (QPENf + D);
    {                                                    // re = A z - 1
      const int j = tid >> 2, l = tid & 3;
      const float* ar = As + j * LDA;
      float acc = 0.f;
      for (int i = l; i < NHID; i += 4) acc += ar[i] * zs[i];
      red[tid] = acc;
    }
    __syncthreads();
    if (tid < NEQ)
      res[tid] = red[4*tid] + red[4*tid+1] + red[4*tid+2] + red[4*tid+3] - 1.0f;
    __syncthreads();

    build_S_wmma(As, invHs, Ss, tid);
    cholesky64(Ss, tid);

    // ---- affine predictor (rs = lam*s) ----
    const float rs_a = lam_i * s_i;
    tops[tid] = -rx + (lam_i * ri - rs_a) / s_i;
    __syncthreads();
    const float dz1 = schur_solve(As, Ss, invHs, tops, res, rvs, red, tid);
    const float ds1 = dz1 - ri;
    const float dl1 = (lam_i * ri - rs_a) / s_i - D * dz1;
    const float aff = fminf(1.0f, 0.999f *
        block_min(fminf(ratf(s_i, ds1), ratf(lam_i, dl1)), red, tid));
    const float mu   = block_sum(s_i * lam_i, red, tid) * (1.0f / NHID);
    const float mu_a = block_sum((s_i + aff * ds1) * (lam_i + aff * dl1),
                                 red, tid) * (1.0f / NHID);
    float sg = mu_a / mu; sg = sg * sg * sg;

    // ---- corrector ----
    const float rs_c = lam_i * s_i + ds1 * dl1 - sg * mu;
    tops[tid] = -rx + (lam_i * ri - rs_c) / s_i;
    __syncthreads();
    const float dz2 = schur_solve(As, Ss, invHs, tops, res, rvs, red, tid);
    const float ds2 = dz2 - ri;
    const float dl2 = (lam_i * ri - rs_c) / s_i - D * dz2;
    const float a = fminf(1.0f, 0.999f *
        block_min(fminf(ratf(s_i, ds2), ratf(lam_i, dl2)), red, tid));

    z_i += a * dz2;
    zs[tid] = z_i;
    if (tid < NEQ) nus[tid] += a * rvs[tid];
    s_i   = fmaxf(s_i + a * ds2, 1e-8f);
    lam_i = fmaxf(lam_i + a * dl2, 1e-8f);
    __syncthreads();
  }

  // ---- classifier head + log_softmax (10 classes) ----
  if (tid < NCLS) {
    float acc = b2[tid];
    const float* wr = W2 + tid * NHID;
    for (int i = 0; i < NHID; ++i) acc += zs[i] * wr[i];
    red[tid] = acc;
  }
  __syncthreads();
  if (tid < NCLS) {
    float mx = -3.0e38f;
    for (int c = 0; c < NCLS; ++c) mx = fmaxf(mx, red[c]);
    float se = 0.f;
    for (int c = 0; c < NCLS; ++c) se += expf(red[c] - mx);
    out[b * NCLS + tid] = red[tid] - mx - logf(se);
  }
}

// ======================================================================
extern "C" void kernel_launch(void* const* d_in, const int* in_sizes, int n_in,
                              void* d_out, int out_size, void* d_ws, size_t ws_size,
                              hipStream_t stream) {
  const float* x  = (const float*)d_in[0];
  const float* W1 = (const float*)d_in[1];
  const float* b1 = (const float*)d_in[2];
  const float* W2 = (const float*)d_in[3];
  const float* b2 = (const float*)d_in[4];
  const float* A  = (const float*)d_in[5];
  float* h   = (float*)d_ws;            // 128x256 fp32 scratch
  float* out = (float*)d_out;           // 128x10 fp32

  (void)hipFuncSetAttribute((const void*)qp_head,
      hipFuncAttributeMaxDynamicSharedMemorySize, (int)SMEM_BYTES);

  gemm_h_relu<<<dim3(NHID / 16, NBATCH / 16), 32, 0, stream>>>(x, W1, b1, h);
  qp_head<<<NBATCH, NHID, SMEM_BYTES, stream>>>(h, A, W2, b2, out);
}